// BillehColumn_59339268162065
// MI455X (gfx1250) — compile-verified
//
#include <hip/hip_runtime.h>
#include <math.h>

// ---------------- Problem constants (from reference) ----------------
#define NN    50000      // neurons
#define RR    4          // receptors
#define DD    5          // delay slots
#define BB    8          // batch
#define TT    10         // timesteps
#define EREC  1500000
#define EIN   200000
#define NIN   17400
#define DT_   1.0f

#define NB    (NN * BB)          // 400,000
#define RNB   (RR * NN * BB)     // 1,600,000

#define TILE  2048               // edges staged per block
#define HALF  1024               // pipeline granularity
#define STHREADS 256

// ---------------- gfx1250 async global->LDS staging ----------------
#ifndef __has_builtin
#define __has_builtin(x) 0
#endif

#if defined(__AMDGCN__) && __has_builtin(__builtin_amdgcn_global_load_async_to_lds_b128)
#define ASYNC_LDS 1
#else
#define ASYNC_LDS 0
#endif

// b128 async copy operates on 4-int vectors; params are AS1 (global / __device__)
// source and AS3 (LDS / __shared__) destination pointers.
typedef int v4i __attribute__((ext_vector_type(4)));
typedef __attribute__((address_space(1))) v4i as1_v4i;
typedef __attribute__((address_space(3))) v4i as3_v4i;

// Stage HALF (1024) 4-byte elements g -> lds: one b128 op per thread.
__device__ __forceinline__ void stage_1024(void* lds, const void* g, int tid) {
#if ASYNC_LDS
  int off = tid * 16;  // bytes
  void* gp = (void*)((const char*)g + off);
  void* lp = (void*)((char*)lds + off);
  __builtin_amdgcn_global_load_async_to_lds_b128((as1_v4i*)gp, (as3_v4i*)lp, 0, 0);
#else
  ((float4*)lds)[tid] = ((const float4*)g)[tid];
#endif
}

// Wait until at most `N` of this wave's async ops remain outstanding.
// N must be a compile-time constant (encodes into s_wait_asynccnt SIMM16).
template <int N>
__device__ __forceinline__ void async_wait() {
#if ASYNC_LDS
#if __has_builtin(__builtin_amdgcn_s_wait_asynccnt)
  __builtin_amdgcn_s_wait_asynccnt(N);
#else
  asm volatile("s_wait_asynccnt %0" :: "i"(N) : "memory");
#endif
#endif
}

// Native FP32 global atomic add (global_atomic_add_f32, RNE, no CAS loop).
__device__ __forceinline__ void atomic_add_f32(float* p, float v) {
#if defined(__AMDGCN__)
  unsafeAtomicAdd(p, v);
#else
  atomicAdd(p, v);
#endif
}

// ---------------- Init: v[n*B+b] = v0[b*N+n] ----------------
__global__ __launch_bounds__(256) void billeh_init_v(float* __restrict__ v,
                                                     const float* __restrict__ v0) {
  int idx = blockIdx.x * blockDim.x + threadIdx.x;
  if (idx >= NB) return;
  int n = idx >> 3;
  int b = idx & 7;
  v[idx] = v0[b * NN + n];
}

// ---------------- Recurrent scatter ----------------
// acc[(tg&3)*NN + (tg>>2)][b] += w * z_hist[slot(d)][src%N][b]
__global__ __launch_bounds__(STHREADS) void billeh_scatter_rec(
    const int* __restrict__ src, const int* __restrict__ tgt,
    const float* __restrict__ w, const float* __restrict__ z_hist,
    float* __restrict__ acc, int E, int t) {
  __shared__ __align__(16) int   s_src[TILE];
  __shared__ __align__(16) int   s_tgt[TILE];
  __shared__ __align__(16) float s_w[TILE];

  const int base = blockIdx.x * TILE;
  const int rem  = E - base;
  const int count = rem < TILE ? rem : TILE;
  const int tid = threadIdx.x;

  if (count == TILE) {
    // --- pipelined async staging: two halves, partial wait between ---
    stage_1024(s_src,        src + base,        tid);   // half 0: 3 ops/wave
    stage_1024(s_tgt,        tgt + base,        tid);
    stage_1024(s_w,          w   + base,        tid);
    stage_1024(s_src + HALF, src + base + HALF, tid);   // half 1: 3 ops/wave
    stage_1024(s_tgt + HALF, tgt + base + HALF, tid);
    stage_1024(s_w   + HALF, w   + base + HALF, tid);
    async_wait<3>();          // half 0 landed (in-order completion)
    __syncthreads();
  } else {
    for (int i = tid; i < count; i += STHREADS) {
      s_src[i] = src[base + i];
      s_tgt[i] = tgt[base + i];
      s_w[i]   = w[base + i];
    }
    __syncthreads();
  }

  const int lim0 = count < HALF ? count : HALF;
  for (int half = 0; half < 2; ++half) {
    int lo = half ? HALF : 0;
    int hi = half ? count : lim0;
    for (int i = lo + tid; i < hi; i += STHREADS) {
      int s = s_src[i];
      int d = s / NN;                    // delay 0..4
      int n = s - d * NN;
      int slot = (t + 9 - d) % DD;       // == (t-1-d) mod 5 ; unwritten slots hold 0
      const float4* zp = (const float4*)(z_hist + ((size_t)slot * NN + n) * BB);
      float4 z0 = zp[0];
      float4 z1 = zp[1];
      // spikes are exactly 0.0 or 1.0 -> sum is an exact nonzero test
      if ((z0.x + z0.y + z0.z + z0.w + z1.x + z1.y + z1.z + z1.w) == 0.0f) continue;
      int tg = s_tgt[i];
      float wv = s_w[i];
      float* ap = acc + (((size_t)(tg & 3)) * NN + (size_t)(tg >> 2)) * BB;
      if (z0.x != 0.0f) atomic_add_f32(ap + 0, wv);
      if (z0.y != 0.0f) atomic_add_f32(ap + 1, wv);
      if (z0.z != 0.0f) atomic_add_f32(ap + 2, wv);
      if (z0.w != 0.0f) atomic_add_f32(ap + 3, wv);
      if (z1.x != 0.0f) atomic_add_f32(ap + 4, wv);
      if (z1.y != 0.0f) atomic_add_f32(ap + 5, wv);
      if (z1.z != 0.0f) atomic_add_f32(ap + 6, wv);
      if (z1.w != 0.0f) atomic_add_f32(ap + 7, wv);
    }
    if (half == 0 && count == TILE) {
      async_wait<0>();        // half 1 landed
      __syncthreads();
    }
  }
}

// ---------------- Input scatter ----------------
// acc[(tg&3)*NN + (tg>>2)][b] += w * x_t[b*NIN + src]
__global__ __launch_bounds__(STHREADS) void billeh_scatter_in(
    const int* __restrict__ src, const int* __restrict__ tgt,
    const float* __restrict__ w, const float* __restrict__ x_t,
    float* __restrict__ acc, int E) {
  __shared__ __align__(16) int   s_src[TILE];
  __shared__ __align__(16) int   s_tgt[TILE];
  __shared__ __align__(16) float s_w[TILE];

  const int base = blockIdx.x * TILE;
  const int rem  = E - base;
  const int count = rem < TILE ? rem : TILE;
  const int tid = threadIdx.x;

  if (count == TILE) {
    stage_1024(s_src,        src + base,        tid);
    stage_1024(s_tgt,        tgt + base,        tid);
    stage_1024(s_w,          w   + base,        tid);
    stage_1024(s_src + HALF, src + base + HALF, tid);
    stage_1024(s_tgt + HALF, tgt + base + HALF, tid);
    stage_1024(s_w   + HALF, w   + base + HALF, tid);
    async_wait<3>();
    __syncthreads();
  } else {
    for (int i = tid; i < count; i += STHREADS) {
      s_src[i] = src[base + i];
      s_tgt[i] = tgt[base + i];
      s_w[i]   = w[base + i];
    }
    __syncthreads();
  }

  const int lim0 = count < HALF ? count : HALF;
  for (int half = 0; half < 2; ++half) {
    int lo = half ? HALF : 0;
    int hi = half ? count : lim0;
    for (int i = lo + tid; i < hi; i += STHREADS) {
      int s  = s_src[i];
      int tg = s_tgt[i];
      float wv = s_w[i];
      float* ap = acc + (((size_t)(tg & 3)) * NN + (size_t)(tg >> 2)) * BB;
#pragma unroll
      for (int b = 0; b < BB; ++b) {
        float xv = x_t[(size_t)b * NIN + s];   // 0.0 or 1.0
        if (xv != 0.0f) atomic_add_f32(ap + b, wv);
      }
    }
    if (half == 0 && count == TILE) {
      async_wait<0>();
      __syncthreads();
    }
  }
}

// ---------------- Neuron dynamics (one thread per (n,b)) ----------------
__global__ __launch_bounds__(256) void billeh_neuron(
    float* __restrict__ v, float* __restrict__ rstate,
    float* __restrict__ a1, float* __restrict__ a2,
    float* __restrict__ psc_rise, float* __restrict__ psc,
    float* __restrict__ acc, float* __restrict__ z_hist,
    const float* __restrict__ bkg_w, const float* __restrict__ decay,
    const float* __restrict__ current_factor, const float* __restrict__ v_th,
    const float* __restrict__ e_l, const float* __restrict__ v_reset,
    const float* __restrict__ t_ref, const float* __restrict__ asc_amps,
    const float* __restrict__ param_k, const float* __restrict__ param_g,
    const float* __restrict__ syn_decay, const float* __restrict__ psc_initial,
    float* __restrict__ out_t, int t) {
  int idx = blockIdx.x * blockDim.x + threadIdx.x;   // idx = n*B + b
  if (idx >= NB) return;
  int n = idx >> 3;
  int b = idx & 7;

  const int slot_prev = (t + 4) % DD;   // (t-1) mod 5 ; zeros at t=0
  const int slot_cur  = t % DD;

  float prev_z = z_hist[(size_t)slot_prev * NB + idx];

  float vv = v[idx];
  float rr = rstate[idx];
  float x1 = a1[idx];
  float x2 = a2[idx];

  float input_cur = 0.0f;
#pragma unroll
  for (int r = 0; r < RR; ++r) {
    size_t off = (size_t)r * NB + idx;
    float sd = syn_decay[n * RR + r];
    float pi = psc_initial[n * RR + r];
    float total = acc[off] + bkg_w[n * RR + r];
    acc[off] = 0.0f;                      // reset accumulator for next step
    float pr = psc_rise[off];
    float pc = psc[off];
    input_cur += pc;                      // uses OLD psc
    psc_rise[off] = sd * pr + pi * total;
    psc[off]      = pc * sd + DT_ * sd * pr;   // uses OLD psc_rise
  }

  float k0 = param_k[n * 2 + 0];
  float k1 = param_k[n * 2 + 1];
  float ad0 = expf(-DT_ * (1.0f / (1.0f + expf(-k0))));
  float ad1 = expf(-DT_ * (1.0f / (1.0f + expf(-k1))));

  float new_r = fmaxf(rr + prev_z * t_ref[n] - DT_, 0.0f);
  float na1 = ad0 * x1 + prev_z * asc_amps[n * 2 + 0];
  float na2 = ad1 * x2 + prev_z * asc_amps[n * 2 + 1];

  float vth = v_th[n];
  float c1 = input_cur + x1 + x2 + param_g[n] * e_l[n];   // OLD a1/a2
  float nv = decay[n] * vv + current_factor[n] * c1 + prev_z * (v_reset[n] - vth);

  // v_scaled = (nv - vth)/(vth - e_l); denominator > 0 so sign(nv - vth) decides
  float nz = (nv > vth) ? 1.0f : 0.0f;
  if (new_r > 0.0f) nz = 0.0f;

  v[idx] = nv;
  rstate[idx] = new_r;
  a1[idx] = na1;
  a2[idx] = na2;
  z_hist[(size_t)slot_cur * NB + idx] = nz;
  out_t[(size_t)b * NN + n] = nz;       // zs[t, b, n]
}

// ---------------- Host-side launch ----------------
extern "C" void kernel_launch(void* const* d_in, const int* in_sizes, int n_in,
                              void* d_out, int out_size, void* d_ws, size_t ws_size,
                              hipStream_t stream) {
  (void)in_sizes; (void)n_in; (void)out_size; (void)ws_size;

  const float* x         = (const float*)d_in[0];   // (T,B,NIN)
  const float* w_rec     = (const float*)d_in[1];
  const int*   rec_src   = (const int*)  d_in[2];
  const int*   rec_tgt   = (const int*)  d_in[3];
  const float* w_in      = (const float*)d_in[4];
  const int*   in_src    = (const int*)  d_in[5];
  const int*   in_tgt    = (const int*)  d_in[6];
  const float* bkg_w     = (const float*)d_in[7];
  const float* decay     = (const float*)d_in[8];
  const float* cfactor   = (const float*)d_in[9];
  const float* v_th      = (const float*)d_in[10];
  const float* e_l       = (const float*)d_in[11];
  const float* v_reset   = (const float*)d_in[12];
  const float* t_ref     = (const float*)d_in[13];
  const float* asc_amps  = (const float*)d_in[14];  // (N,2)
  const float* param_k   = (const float*)d_in[15];  // (N,2)
  const float* param_g   = (const float*)d_in[16];
  const float* syn_decay = (const float*)d_in[17];  // (N,R)
  const float* psc_init  = (const float*)d_in[18];  // (N,R)
  const float* v0        = (const float*)d_in[19];  // (B,N)

  float* out = (float*)d_out;
  float* ws  = (float*)d_ws;

  // Workspace layout (floats)
  float* z_hist   = ws;                  // 2,000,000  [D][N][B]
  float* v        = ws + 2000000;        // NB
  float* rstate   = ws + 2400000;        // NB
  float* a1       = ws + 2800000;        // NB
  float* a2       = ws + 3200000;        // NB
  float* psc_rise = ws + 3600000;        // RNB  [r][N][B]
  float* psc      = ws + 5200000;        // RNB
  float* acc      = ws + 6800000;        // RNB
  const size_t ws_floats = 8400000;      // 33.6 MB

  (void)hipMemsetAsync(d_ws, 0, ws_floats * sizeof(float), stream);
  billeh_init_v<<<(NB + 255) / 256, 256, 0, stream>>>(v, v0);

  const int rec_blocks = (EREC + TILE - 1) / TILE;
  const int in_blocks  = (EIN  + TILE - 1) / TILE;
  const int nrn_blocks = (NB + 255) / 256;

  for (int t = 0; t < TT; ++t) {
    billeh_scatter_rec<<<rec_blocks, STHREADS, 0, stream>>>(
        rec_src, rec_tgt, w_rec, z_hist, acc, EREC, t);
    billeh_scatter_in<<<in_blocks, STHREADS, 0, stream>>>(
        in_src, in_tgt, w_in, x + (size_t)t * BB * NIN, acc, EIN);
    billeh_neuron<<<nrn_blocks, 256, 0, stream>>>(
        v, rstate, a1, a2, psc_rise, psc, acc, z_hist,
        bkg_w, decay, cfactor, v_th, e_l, v_reset, t_ref,
        asc_amps, param_k, param_g, syn_decay, psc_init,
        out + (size_t)t * BB * NN, t);
  }
}